// ResonantHTTEmbedding_77867757076823
// MI455X (gfx1250) — compile-verified
//
#include <hip/hip_runtime.h>
#include <hip/hip_bf16.h>
#include <math.h>

typedef __attribute__((ext_vector_type(2))) float v2f;
typedef __attribute__((ext_vector_type(8))) float v8f;

// ---------------------------------------------------------------------------
// Stage A: table1[hi][e][s1], hi = i0*16+i1, e = D*8+d1  (256 x 64 x 16 f32)
// T1[s1][e] = sum_r (core0[i0,0,r,D] * cos(phase[r])) * core1[i1,r,s1,d1]
// Stored [e][s1] (A-operand row-major M x K) for direct WMMA fragment loads.
// ---------------------------------------------------------------------------
__global__ void build_table1(const float* __restrict__ core0,
                             const float* __restrict__ core1,
                             const float* __restrict__ phase,
                             float* __restrict__ t1) {
    const int hi = blockIdx.x;            // 0..255
    const int i0 = hi >> 4, i1 = hi & 15;
    __shared__ float cp[16];
    if (threadIdx.x < 16) cp[threadIdx.x] = cosf(phase[threadIdx.x]);
    __syncthreads();
    for (int idx = threadIdx.x; idx < 1024; idx += blockDim.x) {
        const int e  = idx >> 4;          // 0..63
        const int s1 = idx & 15;
        const int D  = e >> 3, d1 = e & 7;
        float acc = 0.0f;
#pragma unroll
        for (int r = 0; r < 16; ++r) {
            const float a = core0[i0 * 128 + r * 8 + D] * cp[r];
            const float b = core1[i1 * 2048 + r * 128 + s1 * 8 + d1];
            acc = fmaf(a, b, acc);
        }
        t1[hi * 1024 + e * 16 + s1] = acc;
    }
}

// ---------------------------------------------------------------------------
// Stage B: table2[lo][f][s1], lo = i2*16+i3, f = d2*4+d3  (256 x 16 x 16 f32)
// T2[s1][f] = sum_s2 core2[i2,s1,s2,d2] * core3[i3,s2,0,d3]
// Stored transposed [f][s1] so B fragments load with the same pattern as A.
// ---------------------------------------------------------------------------
__global__ void build_table2(const float* __restrict__ core2,
                             const float* __restrict__ core3,
                             float* __restrict__ t2) {
    const int lo = blockIdx.x;            // 0..255
    const int i2 = lo >> 4, i3 = lo & 15;
    for (int idx = threadIdx.x; idx < 256; idx += blockDim.x) {
        const int f  = idx >> 4;          // 0..15
        const int s1 = idx & 15;
        const int d2 = f >> 2, d3 = f & 3;
        float acc = 0.0f;
#pragma unroll
        for (int s2 = 0; s2 < 16; ++s2)
            acc = fmaf(core2[i2 * 1024 + s1 * 64 + s2 * 4 + d2],
                       core3[i3 * 64 + s2 * 4 + d3], acc);
        t2[lo * 256 + f * 16 + s1] = acc;
    }
}

// ---------------------------------------------------------------------------
// Main kernel: one token per wave32.
// out[t, e*16 + f] = sum_{s1=0..15} T1[hi(t)][e][s1] * T2[lo(t)][f][s1]
// = (64x16) = A(64x16) * B(16x16) via 4 M-tiles x 4 K-steps of
// V_WMMA_F32_16X16X4_F32.  EXEC is all-1s (no divergence by construction).
// ---------------------------------------------------------------------------
__global__ void __launch_bounds__(256)
htt_embed(const int* __restrict__ ids,
          const float* __restrict__ t1,
          const float* __restrict__ t2,
          float* __restrict__ out, int ntok) {
    const int lane   = threadIdx.x & 31;
    const int wave   = (int)((blockIdx.x * blockDim.x + threadIdx.x) >> 5);
    const int nwaves = (int)((gridDim.x * blockDim.x) >> 5);
    const int m    = lane & 15;   // A row / B col / D col
    const int half = lane >> 4;   // K sub-slot selector
    const int koff = half * 2;

    for (int t = wave; t < ntok; t += nwaves) {
        const int id = ids[t];                       // id < 65536
        const float* __restrict__ A = t1 + (size_t)((id >> 8) & 255) * 1024;
        const float* __restrict__ B = t2 + (size_t)(id & 255) * 256;

        // B fragments (shared across all 4 M-tiles): one float2 per K-step.
        v2f bfrag[4];
#pragma unroll
        for (int kk = 0; kk < 4; ++kk)
            bfrag[kk] = *(const v2f*)(B + m * 16 + kk * 4 + koff);

        float* __restrict__ outt = out + (size_t)t * 1024;
#pragma unroll
        for (int et = 0; et < 4; ++et) {
            v8f acc = {0.f, 0.f, 0.f, 0.f, 0.f, 0.f, 0.f, 0.f};
#pragma unroll
            for (int kk = 0; kk < 4; ++kk) {
                const v2f afrag =
                    *(const v2f*)(A + (et * 16 + m) * 16 + kk * 4 + koff);
                // 8 args: (neg_a, A, neg_b, B, c_mod, C, reuse_a, reuse_b)
                acc = __builtin_amdgcn_wmma_f32_16x16x4_f32(
                    false, afrag, false, bfrag[kk], (short)0, acc, false, false);
            }
            // D layout: VGPR v -> row (et*16 + v + 8*half), col m.
#pragma unroll
            for (int v = 0; v < 8; ++v)
                outt[(et * 16 + v + 8 * half) * 16 + m] = acc[v];
        }
    }
}

extern "C" void kernel_launch(void* const* d_in, const int* in_sizes, int n_in,
                              void* d_out, int out_size, void* d_ws, size_t ws_size,
                              hipStream_t stream) {
    const int*   ids   = (const int*)d_in[0];
    const float* core0 = (const float*)d_in[1];
    const float* core1 = (const float*)d_in[2];
    const float* core2 = (const float*)d_in[3];
    const float* core3 = (const float*)d_in[4];
    const float* phase = (const float*)d_in[5];

    float* t1 = (float*)d_ws;              // 256*1024 f32 = 1 MB
    float* t2 = t1 + 256 * 1024;           // 256*256  f32 = 256 KB
    float* out = (float*)d_out;
    const int ntok = in_sizes[0];          // 8 * 2048 = 16384

    build_table1<<<256, 256, 0, stream>>>(core0, core1, phase, t1);
    build_table2<<<256, 256, 0, stream>>>(core2, core3, t2);
    htt_embed<<<512, 256, 0, stream>>>(ids, t1, t2, out, ntok);
}